// WeightedHead_6227702579637
// MI455X (gfx1250) — compile-verified
//
#include <hip/hip_runtime.h>

#define B_  64
#define KH_ 3
#define L_  512
#define DM_ 1024
#define H_  1024
#define BK_ (B_ * KH_)   // 192

typedef __attribute__((ext_vector_type(2))) float v2f;
typedef __attribute__((ext_vector_type(8))) float v8f;

// ---------------------------------------------------------------------------
// Zero the xsum accumulator region (atomic target, must be 0 each call).
// grid 192, block 256, 4 floats/thread -> 192*1024 floats.
__global__ void k_zero_xsum(float* __restrict__ xsum) {
    int idx = (blockIdx.x * 256 + threadIdx.x) * 4;
    *(float4*)(xsum + idx) = make_float4(0.f, 0.f, 0.f, 0.f);
}

// ---------------------------------------------------------------------------
// weights = softmax(W_comb @ graph_feature + b_comb); 1 block, 3 waves.
__global__ void k_weights(const float* __restrict__ Wc,
                          const float* __restrict__ gf,
                          const float* __restrict__ bc,
                          float* __restrict__ w_out) {
    int wv   = threadIdx.x >> 5;   // 0..2 : row of W_comb
    int lane = threadIdx.x & 31;
    float s = 0.f;
    for (int d = lane; d < DM_; d += 32)
        s += Wc[wv * DM_ + d] * gf[d];
    for (int off = 16; off > 0; off >>= 1)
        s += __shfl_xor(s, off, 32);
    __shared__ float lg[KH_];
    if (lane == 0) lg[wv] = s + bc[wv];
    __syncthreads();
    if (threadIdx.x == 0) {
        float m  = fmaxf(lg[0], fmaxf(lg[1], lg[2]));
        float e0 = expf(lg[0] - m), e1 = expf(lg[1] - m), e2 = expf(lg[2] - m);
        float inv = 1.f / (e0 + e1 + e2);
        w_out[0] = e0 * inv; w_out[1] = e1 * inv; w_out[2] = e2 * inv;
    }
}

// ---------------------------------------------------------------------------
// counts[bk] = number of valid (mask==0) positions. grid 192, block 256.
__global__ void k_counts(const unsigned char* __restrict__ mask,
                         float* __restrict__ counts) {
    __shared__ float red[256];
    int bk = blockIdx.x, t = threadIdx.x;
    const unsigned char* mp = mask + bk * L_;
    red[t] = (mp[t] == 0 ? 1.f : 0.f) + (mp[t + 256] == 0 ? 1.f : 0.f);
    __syncthreads();
    for (int s = 128; s > 0; s >>= 1) {
        if (t < s) red[t] += red[t + s];
        __syncthreads();
    }
    if (t == 0) counts[bk] = red[0];
}

// ---------------------------------------------------------------------------
// xsum[bk,d] += sum over a 64-row L-chunk of valid rows of x.
// grid (192, 8), block 256. float4 coalesced streaming; mask staged in LDS so
// the row-skip branch is wave-uniform (skips ~50% of HBM traffic).
__global__ void k_masked_sum(const float* __restrict__ x,
                             const unsigned char* __restrict__ mask,
                             float* __restrict__ xsum) {
    const int bk = blockIdx.x;
    const int l0 = blockIdx.y * 64;
    const int c  = threadIdx.x * 4;

    __shared__ unsigned char mv[64];
    if (threadIdx.x < 64) mv[threadIdx.x] = mask[bk * L_ + l0 + threadIdx.x];
    __syncthreads();

    const float4* xp = (const float4*)(x + ((size_t)bk * L_ + l0) * DM_ + c);
    float4 acc = make_float4(0.f, 0.f, 0.f, 0.f);
    for (int l = 0; l < 64; ++l) {
        __builtin_prefetch(xp + (size_t)(l + 4) * (DM_ / 4), 0, 0); // global_prefetch_b8
        if (mv[l] == 0) {
            float4 v = xp[(size_t)l * (DM_ / 4)];
            acc.x += v.x; acc.y += v.y; acc.z += v.z; acc.w += v.w;
        }
    }
    float* o = xsum + bk * DM_ + c;
    atomicAdd(o + 0, acc.x);
    atomicAdd(o + 1, acc.y);
    atomicAdd(o + 2, acc.z);
    atomicAdd(o + 3, acc.w);
}

// ---------------------------------------------------------------------------
// y[b,d] = sum_k (w[k]/counts[b,k]) * xsum[b,k,d]. grid 64, block 256.
__global__ void k_combine(const float* __restrict__ xsum,
                          const float* __restrict__ w,
                          const float* __restrict__ counts,
                          float* __restrict__ y) {
    int b = blockIdx.x;
    int c = threadIdx.x * 4;
    float s0 = w[0] / counts[b * 3 + 0];
    float s1 = w[1] / counts[b * 3 + 1];
    float s2 = w[2] / counts[b * 3 + 2];
    float4 v0 = *(const float4*)(xsum + (size_t)(b * 3 + 0) * DM_ + c);
    float4 v1 = *(const float4*)(xsum + (size_t)(b * 3 + 1) * DM_ + c);
    float4 v2 = *(const float4*)(xsum + (size_t)(b * 3 + 2) * DM_ + c);
    float4 r;
    r.x = s0 * v0.x + s1 * v1.x + s2 * v2.x;
    r.y = s0 * v0.y + s1 * v1.y + s2 * v2.y;
    r.z = s0 * v0.z + s1 * v1.z + s2 * v2.z;
    r.w = s0 * v0.w + s1 * v1.w + s2 * v2.w;
    *(float4*)(y + (size_t)b * DM_ + c) = r;
}

// ---------------------------------------------------------------------------
// feature[m, h] = sum_d y[m,d] * W_proj[h,d] + b_proj[h]
// One wave per 16x16 tile, V_WMMA_F32_16X16X4_F32, K-loop of 256 x (K=4).
// A (16x4 f32): vgpr j, lane -> M = lane&15, K = (lane>>4)*2 + j
// B (4x16 f32): vgpr j, lane -> N = lane&15, K = (lane>>4)*2 + j (B[k][n]=W[h0+n][k])
// D (16x16 f32): vgpr r, lane -> M = r + 8*(lane>>4), N = lane&15
__global__ void k_gemm_wmma(const float* __restrict__ y,
                            const float* __restrict__ Wp,
                            const float* __restrict__ bias,
                            float* __restrict__ out) {
    const int h0   = blockIdx.x * 16;
    const int m0   = blockIdx.y * 16;
    const int lane = threadIdx.x;
    const int half = lane >> 4;
    const int lid  = lane & 15;

    const float* arow = y  + (size_t)(m0 + lid) * DM_;
    const float* brow = Wp + (size_t)(h0 + lid) * DM_;

    v8f acc = {};
    for (int k0 = 0; k0 < DM_; k0 += 4) {
        const int ka = k0 + half * 2;
        v2f a = *(const v2f*)(arow + ka);
        v2f b = *(const v2f*)(brow + ka);
        acc = __builtin_amdgcn_wmma_f32_16x16x4_f32(
            /*neg_a=*/false, a, /*neg_b=*/false, b,
            /*c_mod=*/(short)0, acc, /*reuse_a=*/false, /*reuse_b=*/false);
    }
    const float bv = bias[h0 + lid];
    #pragma unroll
    for (int r = 0; r < 8; ++r) {
        int m = r + half * 8;
        out[(size_t)(m0 + m) * H_ + h0 + lid] = acc[r] + bv;
    }
}

// ---------------------------------------------------------------------------
extern "C" void kernel_launch(void* const* d_in, const int* in_sizes, int n_in,
                              void* d_out, int out_size, void* d_ws, size_t ws_size,
                              hipStream_t stream) {
    const float*         gf   = (const float*)d_in[0];          // (DM)
    const float*         x    = (const float*)d_in[1];          // (B,3,L,DM)
    const unsigned char* mask = (const unsigned char*)d_in[2];  // (B,3,L) bool
    const float*         Wp   = (const float*)d_in[3];          // (H,DM)
    const float*         bp   = (const float*)d_in[4];          // (H)
    const float*         Wc   = (const float*)d_in[5];          // (3,DM)
    const float*         bc   = (const float*)d_in[6];          // (3)
    float* out = (float*)d_out;                                  // (B,H)

    float* ws        = (float*)d_ws;
    float* w_weights = ws;                       // 3 (padded to 4)
    float* w_counts  = ws + 4;                   // 192
    float* w_xsum    = ws + 4 + BK_;             // 192*1024 (16B-aligned offset)
    float* w_y       = w_xsum + BK_ * DM_;       // 64*1024

    k_zero_xsum<<<dim3(BK_), dim3(256), 0, stream>>>(w_xsum);
    k_weights<<<dim3(1), dim3(96), 0, stream>>>(Wc, gf, bc, w_weights);
    k_counts<<<dim3(BK_), dim3(256), 0, stream>>>(mask, w_counts);
    k_masked_sum<<<dim3(BK_, L_ / 64), dim3(256), 0, stream>>>(x, mask, w_xsum);
    k_combine<<<dim3(B_), dim3(256), 0, stream>>>(w_xsum, w_weights, w_counts, w_y);
    k_gemm_wmma<<<dim3(H_ / 16, B_ / 16), dim3(32), 0, stream>>>(w_y, Wp, bp, out);
}